// Top2Gating_498216206677
// MI455X (gfx1250) — compile-verified
//
#include <hip/hip_runtime.h>

typedef __attribute__((ext_vector_type(2))) float v2f;
typedef __attribute__((ext_vector_type(4))) float v4f;
typedef __attribute__((ext_vector_type(8))) float v8f;

#define BB   4
#define NN   2048
#define DD   1024
#define EE   16
#define CAP  160
#define NTOK (BB * NN)       /* 8192  */
#define TILES (NTOK / 16)    /* 512   */
#define EC   (EE * CAP)      /* 2560  */

// ---- xor-shuffle within each 16-lane half via v_permlane16_b32 (VALU, no LDS)
template <int M>
__device__ __forceinline__ int permxor16i(int v) {
    constexpr unsigned lo =
        (((0 ^ M) & 0xF) << 0)  | (((1 ^ M) & 0xF) << 4)  |
        (((2 ^ M) & 0xF) << 8)  | (((3 ^ M) & 0xF) << 12) |
        (((4 ^ M) & 0xF) << 16) | (((5 ^ M) & 0xF) << 20) |
        (((6 ^ M) & 0xF) << 24) | (((7 ^ M) & 0xF) << 28);
    constexpr unsigned hi =
        (((8  ^ M) & 0xF) << 0)  | (((9  ^ M) & 0xF) << 4)  |
        (((10 ^ M) & 0xF) << 8)  | (((11 ^ M) & 0xF) << 12) |
        (((12 ^ M) & 0xF) << 16) | (((13 ^ M) & 0xF) << 20) |
        (((14 ^ M) & 0xF) << 24) | (((15 ^ M) & 0xF) << 28);
    return __builtin_amdgcn_permlane16(v, v, lo, hi, false, false);
}
template <int M>
__device__ __forceinline__ float permxor16f(float v) {
    return __builtin_bit_cast(float,
        permxor16i<M>(__builtin_bit_cast(int, v)));
}

// ---------------------------------------------------------------- init
__global__ void init_kernel(int* __restrict__ count1) {
    if (threadIdx.x < BB * EE) count1[threadIdx.x] = 0;
}

// ---------------------------------------------------------------- gating (WMMA f32 16x16x4 + fused softmax/top2)
__global__ __launch_bounds__(256) void gating_kernel(
    const float* __restrict__ x, const float* __restrict__ w,
    const float* __restrict__ probsf,
    int* __restrict__ count1, float* __restrict__ psum,
    float* __restrict__ tg1, float* __restrict__ tg2,
    int* __restrict__ ti1, int* __restrict__ ti2, int* __restrict__ tk2)
{
    __shared__ float sW[DD * EE];              // 64 KB of the 320 KB WGP LDS
    const int tid = threadIdx.x;
    for (int i = tid; i < DD * EE; i += 256) sW[i] = w[i];
    __syncthreads();

    const int wave  = tid >> 5;
    const int lane  = tid & 31;
    const int tile  = blockIdx.x * 8 + wave;   // 16 tokens per wave
    const int n     = lane & 15;               // A: row / B: column
    const int khalf = lane >> 4;               // lanes 16-31 hold K+2,K+3

    const float* xA = x + (size_t)(tile * 16 + n) * DD + khalf * 2;

    v8f acc = {};
    #pragma unroll 4
    for (int k0 = 0; k0 < DD; k0 += 4) {
        v2f a = *(const v2f*)(xA + k0);        // K = k0+2*khalf, +1 (contiguous)
        v2f b;
        b.x = sW[(k0 + 2 * khalf    ) * EE + n];
        b.y = sW[(k0 + 2 * khalf + 1) * EE + n];
        acc = __builtin_amdgcn_wmma_f32_16x16x4_f32(
                  false, a, false, b, (short)0, acc, false, false);
    }

    const int batch = tile >> 7;               // 128 tiles per batch
    float myps = 0.0f;
    #pragma unroll
    for (int i = 0; i < 8; ++i) {
        float lg = acc[i];
        int row = tile * 16 + i + 8 * khalf;   // C/D layout: VGPR i -> rows i, i+8

        // softmax max over the 16 experts (permlane16 xor tree)
        float m = lg;
        m = fmaxf(m, permxor16f<1>(m));
        m = fmaxf(m, permxor16f<2>(m));
        m = fmaxf(m, permxor16f<4>(m));
        m = fmaxf(m, permxor16f<8>(m));
        float p = __expf(lg - m);
        float ssum = p;
        ssum += permxor16f<1>(ssum);
        ssum += permxor16f<2>(ssum);
        ssum += permxor16f<4>(ssum);
        ssum += permxor16f<8>(ssum);
        float prob = p / ssum;
        myps += prob;

        // top-1 (tie -> lowest index, matches argmax)
        float v1 = prob; int i1 = n;
        #define AMAX_STAGE(M, V, I)                                        \
            { float ov = permxor16f<M>(V); int oi = permxor16i<M>(I);      \
              if (ov > V || (ov == V && oi < I)) { V = ov; I = oi; } }
        AMAX_STAGE(1, v1, i1) AMAX_STAGE(2, v1, i1)
        AMAX_STAGE(4, v1, i1) AMAX_STAGE(8, v1, i1)
        // top-2 = top-1 of gates with winner zeroed
        float v2 = (n == i1) ? 0.0f : prob; int i2 = n;
        AMAX_STAGE(1, v2, i2) AMAX_STAGE(2, v2, i2)
        AMAX_STAGE(4, v2, i2) AMAX_STAGE(8, v2, i2)
        #undef AMAX_STAGE

        float denom = v1 + v2 + 1e-9f;
        float g1n = v1 / denom;
        float g2n = v2 / denom;
        int kept = probsf[row] < (g2n * 5.0f) ? 1 : 0;   // probs < gate2/0.2

        if (n == 0) {                          // lanes 0 and 16 each own 8 rows
            tg1[row] = g1n; tg2[row] = g2n;
            ti1[row] = i1;  ti2[row] = i2;  tk2[row] = kept;
            atomicAdd(&count1[batch * EE + i1], 1);
        }
    }
    psum[tile * 32 + lane] = myps;             // fixed-order partials for loss
}

// ---------------------------------------------------------------- per-batch serial position scan (wave32, lane-held counters)
__global__ void scan_kernel(
    const int* __restrict__ count1,
    const float* __restrict__ tg1, const float* __restrict__ tg2,
    const int* __restrict__ ti1, const int* __restrict__ ti2,
    const int* __restrict__ tk2,
    int* __restrict__ tslot1, int* __restrict__ tslot2,
    float* __restrict__ tG1, float* __restrict__ tG2)
{
    const int b = blockIdx.x;
    const int lane = threadIdx.x;
    int m1cap = (lane < EE) ? min(count1[b * EE + lane], CAP) : 0;
    int c1 = 0, c2 = 0;
    for (int t = 0; t < NN; ++t) {
        int row = b * NN + t;
        int i1 = ti1[row], i2 = ti2[row], k2 = tk2[row];
        int pos1 = __shfl(c1, i1, 32);
        int pos2 = __shfl(c2, i2, 32) + __shfl(m1cap, i2, 32);
        if (lane == i1) c1++;
        if (k2 && lane == i2) c2++;            // cumsum of pre-capacity mask_2
        if (lane == 0) {
            int m1f = pos1 < CAP;
            int m2f = (k2 && pos2 < CAP) ? 1 : 0;
            // flat slot in the [E, CAP] slab; -1 == no contribution
            tslot1[row] = m1f ? (i1 * CAP + pos1) : -1;
            tslot2[row] = m2f ? (i2 * CAP + pos2) : -1;
            tG1[row] = m1f ? tg1[row] : 0.0f;
            tG2[row] = m2f ? tg2[row] : 0.0f;
        }
    }
}

// ---------------------------------------------------------------- deterministic load-balance loss
__global__ void loss_kernel(const float* __restrict__ psum,
                            const int* __restrict__ count1,
                            float* __restrict__ out_loss)
{
    __shared__ float red[64];
    const int t = threadIdx.x;                 // t = b*16 + e
    const int b = t >> 4, e = t & 15;
    float s = 0.0f;
    for (int j = 0; j < TILES / BB; ++j) {     // fixed order: deterministic
        int wv = b * (TILES / BB) + j;
        s += psum[wv * 32 + e] + psum[wv * 32 + 16 + e];
    }
    red[t] = s * (float)count1[b * EE + e];
    __syncthreads();
    for (int st = 32; st >= 1; st >>= 1) {
        if (t < st) red[t] += red[t + st];
        __syncthreads();
    }
    if (t == 0)
        out_loss[0] = red[0] * (16.0f / (4.0f * 2048.0f * 2048.0f)); // e/(b*n^2)
}

// ---------------------------------------------------------------- output fill: 168 MB of NT float4 streaming stores
__global__ __launch_bounds__(256) void fill_kernel(
    const float* __restrict__ tG1, const float* __restrict__ tG2,
    const int* __restrict__ tslot1, const int* __restrict__ tslot2,
    float* __restrict__ disp, float* __restrict__ comb)
{
    const int row = blockIdx.x;
    const float g1 = tG1[row], g2 = tG2[row];
    const int s1 = tslot1[row], s2 = tslot2[row];   // -1 => never matches
    const size_t base4 = (size_t)row * (EC / 4);
    v4f* __restrict__ c4 = (v4f*)comb + base4;
    v4f* __restrict__ d4 = (v4f*)disp + base4;
    for (int v = threadIdx.x; v < EC / 4; v += 256) {
        v4f z = {};
        if ((s1 >> 2) == v) z[s1 & 3] += g1;
        if ((s2 >> 2) == v) z[s2 & 3] += g2;
        v4f dz;
        dz.x = (z.x != 0.0f) ? 1.0f : 0.0f;
        dz.y = (z.y != 0.0f) ? 1.0f : 0.0f;
        dz.z = (z.z != 0.0f) ? 1.0f : 0.0f;
        dz.w = (z.w != 0.0f) ? 1.0f : 0.0f;
        __builtin_nontemporal_store(z,  c4 + v);
        __builtin_nontemporal_store(dz, d4 + v);
    }
}

// ---------------------------------------------------------------- launcher
extern "C" void kernel_launch(void* const* d_in, const int* in_sizes, int n_in,
                              void* d_out, int out_size, void* d_ws, size_t ws_size,
                              hipStream_t stream)
{
    const float* x  = (const float*)d_in[0];
    const float* w  = (const float*)d_in[1];
    const float* pr = (const float*)d_in[2];

    char* ws = (char*)d_ws;
    size_t off = 0;
    int*   count1 = (int*)  (ws + off); off += 1024;
    float* psum   = (float*)(ws + off); off += (size_t)TILES * 32 * 4;   // 64 KB
    float* tg1 = (float*)(ws + off); off += NTOK * 4;
    float* tg2 = (float*)(ws + off); off += NTOK * 4;
    int*   ti1 = (int*)  (ws + off); off += NTOK * 4;
    int*   ti2 = (int*)  (ws + off); off += NTOK * 4;
    int*   tk2 = (int*)  (ws + off); off += NTOK * 4;
    int*   ts1 = (int*)  (ws + off); off += NTOK * 4;
    int*   ts2 = (int*)  (ws + off); off += NTOK * 4;
    float* tG1 = (float*)(ws + off); off += NTOK * 4;
    float* tG2 = (float*)(ws + off); off += NTOK * 4;

    float* disp  = (float*)d_out;
    float* comb  = disp + (size_t)NTOK * EC;
    float* lossp = comb + (size_t)NTOK * EC;

    init_kernel<<<1, 64, 0, stream>>>(count1);
    gating_kernel<<<TILES / 8, 256, 0, stream>>>(x, w, pr, count1, psum,
                                                 tg1, tg2, ti1, ti2, tk2);
    scan_kernel<<<BB, 32, 0, stream>>>(count1, tg1, tg2, ti1, ti2, tk2,
                                       ts1, ts2, tG1, tG2);
    loss_kernel<<<1, 64, 0, stream>>>(psum, count1, lossp);
    fill_kernel<<<NTOK, 256, 0, stream>>>(tG1, tG2, ts1, ts2, disp, comb);
}